// LayerBlock_64561948393642
// MI455X (gfx1250) — compile-verified
//
#include <hip/hip_runtime.h>
#include <hip/hip_bf16.h>
#include <math.h>

// Problem sizes (fixed by the reference)
#define N_NODES 50000
#define N_EDGES 800000
#define IN_NDIM 128
#define IN_EDIM 64
#define HOUT    128     // H*OUT_N == H*OUT_E == 128
#define BN_EPS  1e-5f

typedef __attribute__((ext_vector_type(2))) float v2f;
typedef __attribute__((ext_vector_type(8))) float v8f;

// Workspace layout (in floats):
//   P      : N*384   (cols 0..127 = h@W_nsrc, 128..255 = h@W_ndst, 256..383 = h@W_node + b_node)
//   scores : E*4     (raw scores, then exp(scores - max))
//   mmax   : N*4     (segment max)
//   den    : N*4     (segment sum of exp)
//   stats  : 512     (esum[128], esumsq[128], nsum[128], nsumsq[128])

// ---------------------------------------------------------------- init
__global__ void k0_init(float* __restrict__ mmax, float* __restrict__ den,
                        float* __restrict__ stats) {
  int i = blockIdx.x * blockDim.x + threadIdx.x;
  if (i < N_NODES * 4) { mmax[i] = -__builtin_inff(); den[i] = 0.f; }
  if (i < 512) stats[i] = 0.f;
}

// --------------------------------------------------- node GEMMs (WMMA f32)
// C(N x 512) = h(N x 128) x [W_nsrc | W_ndst | W_node | W_res], one wave per
// 16x16 tile per matrix. Residual quadrant goes directly to hout (d_out).
__global__ __launch_bounds__(256) void k1_node_gemm(
    const float* __restrict__ h,
    const float* __restrict__ W0, const float* __restrict__ W1,
    const float* __restrict__ W2, const float* __restrict__ W3,
    const float* __restrict__ b_node,
    float* __restrict__ P, float* __restrict__ hout) {
  const int wave = threadIdx.x >> 5;
  const int lane = threadIdx.x & 31;
  const int l15  = lane & 15;
  const int kh   = (lane >> 4) << 1;      // 0 or 2: K sub-offset for this half-wave
  const int row0 = blockIdx.x << 4;
  const int mrow = row0 + l15;            // A row held by this lane
  const int col  = (wave << 4) + l15;     // output column within each 128-wide quadrant

  v8f acc0 = {}, acc1 = {}, acc2 = {}, acc3 = {};
  for (int k = 0; k < IN_NDIM; k += 4) {
    v2f a = *(const v2f*)(h + (size_t)mrow * IN_NDIM + k + kh);
    v2f b0, b1, b2, b3;
    const int r0 = (k + kh) * IN_NDIM + col;
    const int r1 = r0 + IN_NDIM;
    b0.x = W0[r0]; b0.y = W0[r1];
    b1.x = W1[r0]; b1.y = W1[r1];
    b2.x = W2[r0]; b2.y = W2[r1];
    b3.x = W3[r0]; b3.y = W3[r1];
    acc0 = __builtin_amdgcn_wmma_f32_16x16x4_f32(false, a, false, b0, (short)0, acc0, false, false);
    acc1 = __builtin_amdgcn_wmma_f32_16x16x4_f32(false, a, false, b1, (short)0, acc1, false, false);
    acc2 = __builtin_amdgcn_wmma_f32_16x16x4_f32(false, a, false, b2, (short)0, acc2, false, false);
    acc3 = __builtin_amdgcn_wmma_f32_16x16x4_f32(false, a, false, b3, (short)0, acc3, false, false);
  }
  const float bn = b_node[col];
  const int rbase = row0 + ((lane >> 4) << 3);
#pragma unroll
  for (int j = 0; j < 8; ++j) {
    const int r = rbase + j;
    P[(size_t)r * 384 + col]        = acc0[j];
    P[(size_t)r * 384 + 128 + col]  = acc1[j];
    P[(size_t)r * 384 + 256 + col]  = acc2[j] + bn;   // hn = h@W_node + b_node
    hout[(size_t)r * 128 + col]     = acc3[j];        // residual h@W_res
  }
}

// ------------------------------------- edge GEMM + gather epilogue (WMMA f32)
// f = e@W_edge + Psrc[src] + Pdst[dst] + b_e, written straight to the e2
// region of d_out. Block = 16 edges x 128 cols (8 waves x one 16x16 tile).
__global__ __launch_bounds__(256) void k2_edge_gemm(
    const float* __restrict__ e, const float* __restrict__ Wedge,
    const float* __restrict__ b_e,
    const int* __restrict__ src, const int* __restrict__ dst,
    const float* __restrict__ P, float* __restrict__ fout) {
  const int wave = threadIdx.x >> 5;
  const int lane = threadIdx.x & 31;
  const int l15  = lane & 15;
  const int kh   = (lane >> 4) << 1;
  const int row0 = blockIdx.x << 4;
  const int mrow = row0 + l15;
  const int col  = (wave << 4) + l15;

  v8f acc = {};
  for (int k = 0; k < IN_EDIM; k += 4) {
    v2f a = *(const v2f*)(e + (size_t)mrow * IN_EDIM + k + kh);
    v2f b;
    b.x = Wedge[(k + kh) * HOUT + col];
    b.y = Wedge[(k + kh + 1) * HOUT + col];
    acc = __builtin_amdgcn_wmma_f32_16x16x4_f32(false, a, false, b, (short)0, acc, false, false);
  }
  const float bev = b_e[col];
  const int rbase = row0 + ((lane >> 4) << 3);
#pragma unroll
  for (int j = 0; j < 8; ++j) {
    const int r = rbase + j;
    const int s = src[r], d = dst[r];
    float fv = acc[j] + P[(size_t)s * 384 + col] + P[(size_t)d * 384 + 128 + col] + bev;
    fout[(size_t)r * 128 + col] = fv;
  }
}

// ----------------- fused pass over f: scores + segment-max + edge BN stats
// Wave w handles head (w&3) of edge (base + 2*it + (w>>2)); lane == column
// within head, so a full wave reduction (shfl_xor) yields the score.
#define K3_EDGES 128
__global__ __launch_bounds__(256) void k3_scores(
    const float* __restrict__ fout, const float* __restrict__ attn,
    const int* __restrict__ dst, float* __restrict__ scores,
    float* __restrict__ mmax, float* __restrict__ esum, float* __restrict__ esumsq) {
  const int wave = threadIdx.x >> 5;
  const int lane = threadIdx.x & 31;
  const int hh   = wave & 3;
  const int eoff = wave >> 2;           // 2 edges in flight per iteration
  const int col  = hh * 32 + lane;
  const float aw = attn[col];           // attn is (4,32) row-major
  float sacc = 0.f, s2acc = 0.f;
  const int base = blockIdx.x * K3_EDGES;
  for (int it = 0; it < K3_EDGES / 2; ++it) {
    const int edge = base + it * 2 + eoff;
    const float fv = fout[(size_t)edge * 128 + col];
    sacc += fv; s2acc += fv * fv;
    float sc = (fv > 0.f ? fv : 0.2f * fv) * aw;   // leaky_relu(0.2) * attn
#pragma unroll
    for (int off = 16; off > 0; off >>= 1) sc += __shfl_xor(sc, off, 32);
    if (lane == 0) {
      scores[edge * 4 + hh] = sc;
      atomicMax(&mmax[dst[edge] * 4 + hh], sc);    // global_atomic_max_num_f32
    }
  }
  atomicAdd(&esum[col], sacc);
  atomicAdd(&esumsq[col], s2acc);
}

// ------------------------------------------------ exp(scores - max) + denom
__global__ void k4_exp(const int* __restrict__ dst, const float* __restrict__ mmax,
                       float* __restrict__ scores, float* __restrict__ den) {
  const int i = blockIdx.x * blockDim.x + threadIdx.x;
  if (i >= N_EDGES * 4) return;
  const int e_id = i >> 2, hh = i & 3;
  const float ex = __expf(scores[i] - mmax[dst[e_id] * 4 + hh]);
  scores[i] = ex;                               // reuse buffer as numerator
  atomicAdd(&den[dst[e_id] * 4 + hh], ex);
}

// --------------------------- alpha-weighted message scatter (atomic to L2)
__global__ __launch_bounds__(256) void k5_scatter(
    const int* __restrict__ src, const int* __restrict__ dst,
    const float* __restrict__ P, const float* __restrict__ scores,
    const float* __restrict__ den, float* __restrict__ hout) {
  const int t = threadIdx.x;
  const int edge = blockIdx.x * 2 + (t >> 7);
  const int c = t & 127;
  const int hh = c >> 5;
  const int s = src[edge], d = dst[edge];
  const float alpha = scores[edge * 4 + hh] / den[d * 4 + hh];
  const float v = P[(size_t)s * 384 + 256 + c] * alpha;   // hn[src] * alpha
  atomicAdd(&hout[(size_t)d * 128 + c], v);
}

// -------------------------------------------------------- node BN statistics
#define K6_ROWS 64
__global__ __launch_bounds__(256) void k6_nstats(const float* __restrict__ hout,
                                                 float* __restrict__ nsum,
                                                 float* __restrict__ nsumsq) {
  const int t = threadIdx.x;
  const int col = t & 127, half = t >> 7;
  const int base = blockIdx.x * K6_ROWS;
  float s = 0.f, s2 = 0.f;
  for (int r = base + half; r < base + K6_ROWS && r < N_NODES; r += 2) {
    const float v = hout[(size_t)r * 128 + col];
    s += v; s2 += v * v;
  }
  atomicAdd(&nsum[col], s);
  atomicAdd(&nsumsq[col], s2);
}

// -------------------------------------------- node BN + leaky_relu, in place
__global__ void k7_nnorm(const float* __restrict__ nsum, const float* __restrict__ nsumsq,
                         const float* __restrict__ gamma, const float* __restrict__ beta,
                         float* __restrict__ hout) {
  const int i = blockIdx.x * blockDim.x + threadIdx.x;
  if (i >= N_NODES * 128) return;
  const int col = i & 127;
  const float mu  = nsum[col] * (1.f / N_NODES);
  const float var = nsumsq[col] * (1.f / N_NODES) - mu * mu;
  const float y = (hout[i] - mu) * rsqrtf(var + BN_EPS) * gamma[col] + beta[col];
  hout[i] = y > 0.f ? y : 0.01f * y;
}

// ------------------------------------------------------ edge BN, in place
__global__ void k8_enorm(const float* __restrict__ esum, const float* __restrict__ esumsq,
                         const float* __restrict__ gamma, const float* __restrict__ beta,
                         float* __restrict__ fout) {
  const size_t i = (size_t)blockIdx.x * blockDim.x + threadIdx.x;
  if (i >= (size_t)N_EDGES * 128) return;
  const int col = (int)(i & 127);
  const float mu  = esum[col] * (1.f / N_EDGES);
  const float var = esumsq[col] * (1.f / N_EDGES) - mu * mu;
  fout[i] = (fout[i] - mu) * rsqrtf(var + BN_EPS) * gamma[col] + beta[col];
}

extern "C" void kernel_launch(void* const* d_in, const int* in_sizes, int n_in,
                              void* d_out, int out_size, void* d_ws, size_t ws_size,
                              hipStream_t stream) {
  const float* h      = (const float*)d_in[0];
  const float* e      = (const float*)d_in[1];
  const int*   src    = (const int*)d_in[2];
  const int*   dst    = (const int*)d_in[3];
  const float* Wnsrc  = (const float*)d_in[4];
  const float* Wndst  = (const float*)d_in[5];
  const float* Wedge  = (const float*)d_in[6];
  const float* b_e    = (const float*)d_in[7];
  const float* attn   = (const float*)d_in[8];
  const float* Wnode  = (const float*)d_in[9];
  const float* b_node = (const float*)d_in[10];
  const float* Wres   = (const float*)d_in[11];
  const float* gam_n  = (const float*)d_in[12];
  const float* bet_n  = (const float*)d_in[13];
  const float* gam_e  = (const float*)d_in[14];
  const float* bet_e  = (const float*)d_in[15];

  float* hout = (float*)d_out;                         // (N, 128)  -> h2
  float* fout = (float*)d_out + (size_t)N_NODES * 128; // (E, 128)  -> f then e2

  float* w      = (float*)d_ws;
  float* P      = w;                                   // N*384
  float* scores = P + (size_t)N_NODES * 384;           // E*4
  float* mmax   = scores + (size_t)N_EDGES * 4;        // N*4
  float* den    = mmax + (size_t)N_NODES * 4;          // N*4
  float* stats  = den + (size_t)N_NODES * 4;           // 512
  float* esum   = stats;
  float* esumsq = stats + 128;
  float* nsum   = stats + 256;
  float* nsumsq = stats + 384;

  k0_init<<<(N_NODES * 4 + 255) / 256, 256, 0, stream>>>(mmax, den, stats);
  k1_node_gemm<<<N_NODES / 16, 256, 0, stream>>>(h, Wnsrc, Wndst, Wnode, Wres,
                                                 b_node, P, hout);
  k2_edge_gemm<<<N_EDGES / 16, 256, 0, stream>>>(e, Wedge, b_e, src, dst, P, fout);
  k3_scores<<<N_EDGES / K3_EDGES, 256, 0, stream>>>(fout, attn, dst, scores,
                                                    mmax, esum, esumsq);
  k4_exp<<<(N_EDGES * 4 + 255) / 256, 256, 0, stream>>>(dst, mmax, scores, den);
  k5_scatter<<<N_EDGES / 2, 256, 0, stream>>>(src, dst, P, scores, den, hout);
  k6_nstats<<<(N_NODES + K6_ROWS - 1) / K6_ROWS, 256, 0, stream>>>(hout, nsum, nsumsq);
  k7_nnorm<<<(N_NODES * 128 + 255) / 256, 256, 0, stream>>>(nsum, nsumsq, gam_n, bet_n, hout);
  k8_enorm<<<(int)(((size_t)N_EDGES * 128 + 255) / 256), 256, 0, stream>>>(esum, esumsq,
                                                                           gam_e, bet_e, fout);
}